// SeastarGATLayer_51969104281683
// MI455X (gfx1250) — compile-verified
//
#include <hip/hip_runtime.h>
#include <math.h>

#define N_NODES 50000
#define E_EDGES 800000
#define IN_F    256
#define HEADS   8
#define OUT_F   32
#define HF      256            // HEADS * OUT_F
#define SLOPE   0.2f

typedef float v2f __attribute__((ext_vector_type(2)));
typedef float v8f __attribute__((ext_vector_type(8)));

// ---------------------------------------------------------------------------
// Kernel 1: feat_src = feat @ fc_w  via V_WMMA_F32_16X16X4_F32
// Block: 256 threads = 8 waves. Block owns 16 rows of A (staged in LDS) and
// all 256 output columns; wave w computes N-tiles {2w, 2w+1}.
// N = 50000 = 3125 * 16 exactly -> no row guard needed.
// ---------------------------------------------------------------------------
__global__ __launch_bounds__(256) void gat_gemm_wmma(
    const float* __restrict__ A,   // [N, 256]
    const float* __restrict__ W,   // [256, 256]
    float* __restrict__ C)         // [N, 256]
{
    __shared__ float As[16 * IN_F];          // 16 KB
    const int block_row = blockIdx.x * 16;
    const int tid  = threadIdx.x;

    // Cooperative coalesced load of the 16x256 A tile (1024 float4's).
    const float4* Ag = (const float4*)(A + (size_t)block_row * IN_F);
    float4* As4 = (float4*)As;
    #pragma unroll
    for (int i = 0; i < 4; ++i)
        As4[tid + i * 256] = Ag[tid + i * 256];
    __syncthreads();

    const int wave  = tid >> 5;
    const int lane  = tid & 31;
    const int lrow  = lane & 15;             // A row / B,C column within tile
    const int khalf = (lane >> 4) << 1;      // 0 (lanes 0-15) or 2 (lanes 16-31)

    #pragma unroll
    for (int nt = 0; nt < 2; ++nt) {
        const int n_base = (wave * 2 + nt) * 16;
        v8f c = {};
        for (int k = 0; k < IN_F; k += 4) {
            v2f a, b;
            // A 16x4 fp32 fragment (ISA layout): lane half selects K pair.
            a[0] = As[lrow * IN_F + k + khalf];
            a[1] = As[lrow * IN_F + k + khalf + 1];
            // B 4x16 fragment: row K striped across lanes within a VGPR.
            b[0] = W[(size_t)(k + khalf)     * HF + n_base + lrow];
            b[1] = W[(size_t)(k + khalf + 1) * HF + n_base + lrow];
            c = __builtin_amdgcn_wmma_f32_16x16x4_f32(
                    false, a, false, b, (short)0, c, false, false);
        }
        // C/D layout: VGPR v -> row v (lanes 0-15) / row v+8 (lanes 16-31).
        const int rbase = block_row + ((lane >> 4) << 3);
        const int col   = n_base + lrow;
        #pragma unroll
        for (int v = 0; v < 8; ++v)
            C[(size_t)(rbase + v) * HF + col] = c[v];
    }
}

// ---------------------------------------------------------------------------
// Kernel 2: el[n,h] = <feat_src[n,h,:], attn_l[h,:]>, same for er.
// One thread per (n,h).
// ---------------------------------------------------------------------------
__global__ __launch_bounds__(256) void gat_scores(
    const float* __restrict__ fs,      // [N, 256]
    const float* __restrict__ attn_l,  // [8, 32]
    const float* __restrict__ attn_r,  // [8, 32]
    float* __restrict__ el,            // [N, 8]
    float* __restrict__ er)            // [N, 8]
{
    int idx = blockIdx.x * blockDim.x + threadIdx.x;
    if (idx >= N_NODES * HEADS) return;
    const int h = idx & (HEADS - 1);
    const int n = idx >> 3;
    const float* f  = fs + (size_t)n * HF + h * OUT_F;
    const float* wl = attn_l + h * OUT_F;
    const float* wr = attn_r + h * OUT_F;
    float sl = 0.f, sr = 0.f;
    #pragma unroll
    for (int j = 0; j < OUT_F; ++j) {
        float v = f[j];
        sl = fmaf(v, wl[j], sl);
        sr = fmaf(v, wr[j], sr);
    }
    el[idx] = sl;
    er[idx] = sr;
}

// ---------------------------------------------------------------------------
// Kernel 3: CSR row_ptr from sorted dst via per-node binary search.
// ---------------------------------------------------------------------------
__global__ __launch_bounds__(256) void gat_row_ptr(
    const int* __restrict__ dst, int* __restrict__ rp)
{
    int n = blockIdx.x * blockDim.x + threadIdx.x;
    if (n > N_NODES) return;
    if (n == N_NODES) { rp[n] = E_EDGES; return; }
    int lo = 0, hi = E_EDGES;                 // lower_bound(dst, n)
    while (lo < hi) {
        int mid = (lo + hi) >> 1;
        if (dst[mid] < n) lo = mid + 1; else hi = mid;
    }
    rp[n] = lo;
}

// ---------------------------------------------------------------------------
// Kernel 4: per-destination edge softmax + weighted gather-sum.
// One block (256 thr = 8 waves) per destination node. wave = head, lane = feat.
// Pass 1 (per head, 32 lanes strided over edges): segment max, then sum of
// exp(leaky_relu(e - max)). Pass 2: every (h,f) thread walks the edge list,
// gather of feat_src[src] is a contiguous 128B line per (edge, head) -> fully
// coalesced; this is the 819 MB roofline stream.
// ---------------------------------------------------------------------------
__global__ __launch_bounds__(256) void gat_aggregate(
    const float* __restrict__ fs,     // [N, 256]
    const float* __restrict__ el,     // [N, 8]
    const float* __restrict__ er,     // [N, 8]
    const int*   __restrict__ src,    // [E]
    const int*   __restrict__ rp,     // [N+1]
    float* __restrict__ out)          // [N, 8, 32]
{
    const int n     = blockIdx.x;
    const int start = rp[n];
    const int end   = rp[n + 1];
    const int tid   = threadIdx.x;
    const int h     = tid >> 5;     // wave index = head
    const int f     = tid & 31;     // lane index = output feature

    __shared__ float m_s[HEADS];
    __shared__ float s_s[HEADS];

    const float ern = er[n * HEADS + h];

    // --- pass 1: max and exp-sum for this head (32 lanes cooperate) ---
    {
        float mx = -INFINITY;
        for (int j = start + f; j < end; j += 32)
            mx = fmaxf(mx, el[src[j] * HEADS + h] + ern);
        #pragma unroll
        for (int off = 16; off; off >>= 1)
            mx = fmaxf(mx, __shfl_xor(mx, off, 32));

        float sum = 0.f;
        for (int j = start + f; j < end; j += 32) {
            float z = el[src[j] * HEADS + h] + ern - mx;
            sum += expf(z >= 0.f ? z : SLOPE * z);
        }
        #pragma unroll
        for (int off = 16; off; off >>= 1)
            sum += __shfl_xor(sum, off, 32);

        if (f == 0) { m_s[h] = mx; s_s[h] = sum; }
    }
    __syncthreads();

    const float m    = m_s[h];
    const float s    = s_s[h];
    const float sinv = (s > 0.f) ? (1.f / s) : 0.f;

    float acc = 0.f;
    for (int j = start; j < end; ++j) {
        const int sn = src[j];
        float z  = el[sn * HEADS + h] + ern - m;     // broadcast load per wave
        float cf = expf(z >= 0.f ? z : SLOPE * z);
        acc = fmaf(cf, fs[(size_t)sn * HF + h * OUT_F + f], acc);
    }
    out[(size_t)n * HF + h * OUT_F + f] = acc * sinv;
}

// ---------------------------------------------------------------------------
extern "C" void kernel_launch(void* const* d_in, const int* in_sizes, int n_in,
                              void* d_out, int out_size, void* d_ws, size_t ws_size,
                              hipStream_t stream)
{
    (void)in_sizes; (void)n_in; (void)out_size; (void)ws_size;

    const float* feat   = (const float*)d_in[0];   // [N, 256]
    const float* fc_w   = (const float*)d_in[1];   // [256, 256]
    const float* attn_l = (const float*)d_in[2];   // [8, 32]
    const float* attn_r = (const float*)d_in[3];   // [8, 32]
    const int*   src    = (const int*)d_in[4];     // [E]
    const int*   dst    = (const int*)d_in[5];     // [E] (sorted)
    float* out = (float*)d_out;                    // [N, 8, 32]

    // Workspace layout (floats): feat_src | el | er | row_ptr(int)
    float* fs = (float*)d_ws;                              // 12,800,000 f
    float* el = fs + (size_t)N_NODES * HF;                 //    400,000 f
    float* er = el + (size_t)N_NODES * HEADS;              //    400,000 f
    int*   rp = (int*)(er + (size_t)N_NODES * HEADS);      //     50,001 i

    // 1) projection GEMM via fp32 WMMA
    gat_gemm_wmma<<<N_NODES / 16, 256, 0, stream>>>(feat, fc_w, fs);

    // 2) attention logits el/er
    {
        int total = N_NODES * HEADS;
        gat_scores<<<(total + 255) / 256, 256, 0, stream>>>(fs, attn_l, attn_r, el, er);
    }

    // 3) CSR row pointers from sorted dst
    gat_row_ptr<<<(N_NODES + 1 + 255) / 256, 256, 0, stream>>>(dst, rp);

    // 4) segmented softmax + weighted gather-sum (bandwidth-bound phase)
    gat_aggregate<<<N_NODES, 256, 0, stream>>>(fs, el, er, src, rp, out);
}